// NetCostGNN_49606872269111
// MI455X (gfx1250) — compile-verified
//
#include <hip/hip_runtime.h>
#include <hip/hip_bf16.h>

#define NNODES 50000
#define NEDGES 800000
#define DIM    128

typedef float v2f __attribute__((ext_vector_type(2)));
typedef float v8f __attribute__((ext_vector_type(8)));

// ---------------------------------------------------------------- utilities

__global__ __launch_bounds__(256) void k_zero_f4(float4* __restrict__ p, int n4) {
    int i = blockIdx.x * blockDim.x + threadIdx.x;
    int stride = gridDim.x * blockDim.x;
    for (; i < n4; i += stride) p[i] = make_float4(0.f, 0.f, 0.f, 0.f);
}

__global__ __launch_bounds__(256) void k_degree(const int* __restrict__ dst,
                                                float* __restrict__ deg, int E) {
    int e = blockIdx.x * blockDim.x + threadIdx.x;
    if (e < E) atomicAdd(&deg[dst[e]], 1.0f);
}

__global__ __launch_bounds__(256) void k_invdeg(float* __restrict__ deg, int n) {
    int i = blockIdx.x * blockDim.x + threadIdx.x;
    if (i < n) deg[i] = 1.0f / fmaxf(deg[i], 1.0f);
}

// ------------------------------------------------------- edge scatter (mean numerator)
// One wave32 per edge: 32 lanes x float4 == 128 features. Row read is one
// coalesced 512B burst; the 128 atomic adds per wave hit contiguous addresses
// of a single dst row (no intra-wave conflicts). agg + h both L2-resident
// (25.6MB each vs 192MB L2), so this phase is L2-atomic bound, not HBM bound.

__global__ __launch_bounds__(256) void k_scatter(const float* __restrict__ h,
                                                 const int* __restrict__ src,
                                                 const int* __restrict__ dst,
                                                 float* __restrict__ agg, int E) {
    int tid = blockIdx.x * blockDim.x + threadIdx.x;
    int e = tid >> 5;
    if (e >= E) return;
    int c = (tid & 31) * 4;
    int s = src[e];
    int d = dst[e];
    const float4 v = *(const float4*)(h + (long)s * DIM + c);
    float* p = agg + (long)d * DIM + c;
    atomicAdd(p + 0, v.x);
    atomicAdd(p + 1, v.y);
    atomicAdd(p + 2, v.z);
    atomicAdd(p + 3, v.w);
}

// ------------------------------------------------- fused SAGE layer via fp32 WMMA
// out = relu( (agg*invdeg) @ Wl + h @ Wr + bias ), fp32 end-to-end.
//
// Register blocking: block = 32-row tile, 8 waves = 8 col tiles of 16.
// Each wave holds TWO 16x16 f32 accumulators (rows m0..15 and m0+16..31)
// so one weight-fragment load (bl/br) feeds two WMMAs -> 4 WMMAs per
// K-step on 12 loaded dwords (was 2 per 8).
//
// Tail (50000 = 32*1562 + 16): only tile 1 of the last block is OOB. Its
// A-loads are clamped to row N-1 (branch-free), stores are predicated.
// No branches before/around the WMMAs, so EXEC stays all-1s as required.
//
// Fragment layouts (ISA 7.12.2, fp32 16x16x4):
//  A 16x4: lanes 0-15 -> M=lane, K={0,1}; lanes 16-31 -> M=lane-16, K={2,3}
//  B 4x16: N striped across lanes; lanes 0-15 -> K={0,1}; 16-31 -> K={2,3}
//  C/D 16x16: VGPR i: lane<16 -> M=i, N=lane; lane>=16 -> M=i+8, N=lane-16

__global__ __launch_bounds__(256) void k_sage_gemm(const float* __restrict__ agg,
                                                   const float* __restrict__ invdeg,
                                                   const float* __restrict__ h,
                                                   const float* __restrict__ Wl,
                                                   const float* __restrict__ Wr,
                                                   const float* __restrict__ bias,
                                                   float* __restrict__ out) {
    const int lane = threadIdx.x & 31;
    const int wave = threadIdx.x >> 5;          // col tile 0..7
    const int m0   = blockIdx.x * 32;
    const int n0   = wave * 16;
    const int half = lane >> 4;                 // 0: K base 0 | 1: K base 2
    const int l16  = lane & 15;
    const int bcol = n0 + l16;                  // B-matrix col owned by lane

    const int r0 = m0 + l16;                            // tile 0 row (< N always)
    const int r1 = min(m0 + 16 + l16, NNODES - 1);      // tile 1 row, clamped

    const float idg0 = invdeg[r0];
    const float idg1 = invdeg[r1];
    const float* agg0 = agg + (long)r0 * DIM + half * 2;
    const float* agg1 = agg + (long)r1 * DIM + half * 2;
    const float* hR0  = h   + (long)r0 * DIM + half * 2;
    const float* hR1  = h   + (long)r1 * DIM + half * 2;

    v8f c0 = {};
    v8f c1 = {};
    #pragma unroll 4
    for (int k = 0; k < DIM; k += 4) {
        // B fragments: loaded ONCE, feed both row tiles
        const int krow = k + half * 2;
        v2f bl, br;
        bl.x = Wl[(long)krow * DIM + bcol];
        bl.y = Wl[(long)(krow + 1) * DIM + bcol];
        br.x = Wr[(long)krow * DIM + bcol];
        br.y = Wr[(long)(krow + 1) * DIM + bcol];

        // A fragments for both tiles (contiguous float pair per lane)
        v2f am0, am1, ah0, ah1;
        am0.x = agg0[k] * idg0;  am0.y = agg0[k + 1] * idg0;
        am1.x = agg1[k] * idg1;  am1.y = agg1[k + 1] * idg1;
        ah0.x = hR0[k];          ah0.y = hR0[k + 1];
        ah1.x = hR1[k];          ah1.y = hR1[k + 1];

        c0 = __builtin_amdgcn_wmma_f32_16x16x4_f32(false, am0, false, bl,
                                                   (short)0, c0, false, false);
        c1 = __builtin_amdgcn_wmma_f32_16x16x4_f32(false, am1, false, bl,
                                                   (short)0, c1, false, false);
        c0 = __builtin_amdgcn_wmma_f32_16x16x4_f32(false, ah0, false, br,
                                                   (short)0, c0, false, false);
        c1 = __builtin_amdgcn_wmma_f32_16x16x4_f32(false, ah1, false, br,
                                                   (short)0, c1, false, false);
    }

    const float bv = bias[bcol];
    #pragma unroll
    for (int i = 0; i < 8; ++i) {
        const int m = m0 + i + half * 8;                 // tile 0: always in range
        out[(long)m * DIM + bcol] = fmaxf(c0[i] + bv, 0.0f);
        const int m2 = m + 16;                           // tile 1: predicated tail
        if (m2 < NNODES)
            out[(long)m2 * DIM + bcol] = fmaxf(c1[i] + bv, 0.0f);
    }
}

// ------------------------------------------------- final projection to scalar
// One wave per node: float4 partial dot + full-wave shuffle reduction (wave32).

__global__ __launch_bounds__(256) void k_final(const float* __restrict__ h,
                                               const float* __restrict__ Wlin,
                                               const float* __restrict__ blin,
                                               float* __restrict__ out, int n) {
    const int lane = threadIdx.x & 31;
    const int node = blockIdx.x * 8 + (threadIdx.x >> 5);
    if (node >= n) return;
    const float4 a = ((const float4*)(h + (long)node * DIM))[lane];
    const float4 w = ((const float4*)Wlin)[lane];
    float s = a.x * w.x + a.y * w.y + a.z * w.z + a.w * w.w;
    #pragma unroll
    for (int off = 16; off > 0; off >>= 1) s += __shfl_down(s, off, 32);
    if (lane == 0) out[node] = s + blin[0];
}

// ---------------------------------------------------------------- launcher

extern "C" void kernel_launch(void* const* d_in, const int* in_sizes, int n_in,
                              void* d_out, int out_size, void* d_ws, size_t ws_size,
                              hipStream_t stream) {
    const float* x     = (const float*)d_in[0];
    const int*   eidx  = (const int*)d_in[1];
    const float* W1l   = (const float*)d_in[2];
    const float* W1r   = (const float*)d_in[3];
    const float* b1    = (const float*)d_in[4];
    const float* W2l   = (const float*)d_in[5];
    const float* W2r   = (const float*)d_in[6];
    const float* b2    = (const float*)d_in[7];
    const float* Wlin  = (const float*)d_in[8];
    const float* blin  = (const float*)d_in[9];

    const int* src = eidx;            // edge_index[0]
    const int* dst = eidx + NEDGES;   // edge_index[1]

    // workspace carve: invdeg[N] | agg[N*128] | h1[N*128]
    float* invdeg = (float*)d_ws;
    float* agg    = invdeg + NNODES;
    float* h1     = agg + (size_t)NNODES * DIM;

    // output tuple: out[N] then h2[N*128]
    float* out_s = (float*)d_out;
    float* h2    = out_s + NNODES;

    const int feat4 = (NNODES * DIM) / 4;                 // float4 count
    const int scatterBlocks = (NEDGES * 32) / 256;        // 100000
    const int gemmBlocks = (NNODES + 31) / 32;            // 1563 (tail-clamped)
    const int finalBlocks = (NNODES + 7) / 8;             // 6250

    // degrees (shared by both layers)
    k_zero_f4<<<(NNODES / 4 + 255) / 256, 256, 0, stream>>>((float4*)invdeg, NNODES / 4);
    k_degree<<<(NEDGES + 255) / 256, 256, 0, stream>>>(dst, invdeg, NEDGES);
    k_invdeg<<<(NNODES + 255) / 256, 256, 0, stream>>>(invdeg, NNODES);

    // layer 1
    k_zero_f4<<<4096, 256, 0, stream>>>((float4*)agg, feat4);
    k_scatter<<<scatterBlocks, 256, 0, stream>>>(x, src, dst, agg, NEDGES);
    k_sage_gemm<<<gemmBlocks, 256, 0, stream>>>(agg, invdeg, x, W1l, W1r, b1, h1);

    // layer 2
    k_zero_f4<<<4096, 256, 0, stream>>>((float4*)agg, feat4);
    k_scatter<<<scatterBlocks, 256, 0, stream>>>(h1, src, dst, agg, NEDGES);
    k_sage_gemm<<<gemmBlocks, 256, 0, stream>>>(agg, invdeg, h1, W2l, W2r, b2, h2);

    // output head
    k_final<<<finalBlocks, 256, 0, stream>>>(h2, Wlin, blin, out_s, NNODES);
}